// Sampler_37383395344474
// MI455X (gfx1250) — compile-verified
//
#include <hip/hip_runtime.h>
#include <stdint.h>

namespace {
constexpr int   kB       = 256;              // batch rows
constexpr int   kV       = 128000;           // vocab
constexpr int   kTpb     = 256;              // threads per block (8 wave32s)
constexpr int   kChunk   = kTpb * 4;         // 1024 floats per chunk per tensor
constexpr int   kNChunks = kV / kChunk;      // 125 (exact)
constexpr int   kRing    = 8;                // async pipeline depth (chunks in flight)
constexpr float kEps     = 1e-10f;
constexpr uint32_t kSlotBytes = kTpb * 16u;  // 4096 B per ring slot
}

// argmax-style comparison: larger value wins, ties -> lower index (jnp.argmax)
__device__ __forceinline__ bool bet(float a, int ai, float b, int bi) {
  return (a > b) || ((a == b) && (ai < bi));
}

// Insert (z,i,s) into a sorted top-3-by-z candidate set held in registers.
__device__ __forceinline__ void ins3(float z, int i, float s,
                                     float& z0, int& i0, float& s0,
                                     float& z1, int& i1, float& s1,
                                     float& z2, int& i2, float& s2) {
  if (!bet(z, i, z2, i2)) return;          // hot path: single guarded exit
  if (bet(z, i, z0, i0)) {
    z2 = z1; i2 = i1; s2 = s1;
    z1 = z0; i1 = i0; s1 = s0;
    z0 = z;  i0 = i;  s0 = s;
  } else if (bet(z, i, z1, i1)) {
    z2 = z1; i2 = i1; s2 = s1;
    z1 = z;  i1 = i;  s1 = s;
  } else {
    z2 = z;  i2 = i;  s2 = s;
  }
}

// Insert (s,i) into a sorted top-4-by-s candidate set held in registers.
__device__ __forceinline__ void ins4(float s, int i,
                                     float& s0, int& i0, float& s1, int& i1,
                                     float& s2, int& i2, float& s3, int& i3) {
  if (!bet(s, i, s3, i3)) return;          // hot path: single guarded exit
  if (bet(s, i, s0, i0)) {
    s3 = s2; i3 = i2; s2 = s1; i2 = i1; s1 = s0; i1 = i0; s0 = s; i0 = i;
  } else if (bet(s, i, s1, i1)) {
    s3 = s2; i3 = i2; s2 = s1; i2 = i1; s1 = s; i1 = i;
  } else if (bet(s, i, s2, i2)) {
    s3 = s2; i3 = i2; s2 = s; i2 = i;
  } else {
    s3 = s; i3 = i;
  }
}

// CDNA5 async global -> LDS copy, 16 bytes per lane, GVS addressing
// (SGPR 64-bit base + per-lane 32-bit byte offset). Tracked by ASYNCcnt.
__device__ __forceinline__ void async_copy_b128(uint32_t lds_byte_addr,
                                                const float* saddr_base,
                                                uint32_t vgpr_byte_off) {
  asm volatile("global_load_async_to_lds_b128 %0, %1, %2"
               :
               : "v"(lds_byte_addr), "v"(vgpr_byte_off), "s"(saddr_base)
               : "memory");
}

__global__ __launch_bounds__(kTpb)
void sampler_fused(const float* __restrict__ logits,
                   const float* __restrict__ temps,
                   const float* __restrict__ phi,
                   const float* __restrict__ noise,
                   const unsigned char* __restrict__ is_tree_p,
                   int* __restrict__ out) {
  // 8-deep async streaming ring (64 KB) overlaid with reduction scratch
  // (~17 KB) -- they are never live at the same time.
  __shared__ union SMem {
    struct {
      float4 L[kRing][kTpb];
      float4 N[kRing][kTpb];
    } ring;
    struct {
      float rz [kTpb][3];
      int   rzi[kTpb][3];
      float rzs[kTpb][3];
      float rs [kTpb][4];
      int   rsi[kTpb][4];
    } red;
  } smem;

  const int t   = threadIdx.x;
  const int row = blockIdx.x;
  const float* rowL = logits + (size_t)row * kV;
  const float* rowN = noise  + (size_t)row * kV;

  const float temp = temps[row];
  const float invT = (temp == 0.0f) ? 1.0f : (1.0f / temp);

  // Generic LDS pointers: low 32 bits == workgroup-relative LDS byte address.
  const uint32_t ldsL = (uint32_t)(uintptr_t)&smem.ring.L[0][0] + (uint32_t)t * 16u;
  const uint32_t ldsN = (uint32_t)(uintptr_t)&smem.ring.N[0][0] + (uint32_t)t * 16u;
  const uint32_t toff = (uint32_t)t * 16u;

  const float NEG = -__builtin_inff();
  const int   BIG = 0x7fffffff;
  // top-3 by logit value (with accompanying log-score)
  float az0 = NEG, az1 = NEG, az2 = NEG;
  int   ai0 = BIG, ai1 = BIG, ai2 = BIG;
  float as0 = NEG, as1 = NEG, as2 = NEG;
  // top-4 by (unpenalized) log-score
  float bs0 = NEG, bs1 = NEG, bs2 = NEG, bs3 = NEG;
  int   bi0 = BIG, bi1 = BIG, bi2 = BIG, bi3 = BIG;

#define PROCESS_ELEM(ZV, EV, IDX)                                              \
  do {                                                                         \
    const float z_ = (ZV);                                                     \
    const float s_ = z_ * invT - __logf((EV) + kEps);                          \
    ins3(z_, (IDX), s_, az0, ai0, as0, az1, ai1, as1, az2, ai2, as2);          \
    ins4(s_, (IDX), bs0, bi0, bs1, bi1, bs2, bi2, bs3, bi3);                   \
  } while (0)

#define PROCESS4(LZ, LE, BASE)                                                 \
  do {                                                                         \
    PROCESS_ELEM((LZ).x, (LE).x, (BASE) + 0);                                  \
    PROCESS_ELEM((LZ).y, (LE).y, (BASE) + 1);                                  \
    PROCESS_ELEM((LZ).z, (LE).z, (BASE) + 2);                                  \
    PROCESS_ELEM((LZ).w, (LE).w, (BASE) + 3);                                  \
  } while (0)

  // Prologue: fill the ring -- 16 outstanding async ops (ASYNCcnt is 6-bit).
#pragma unroll
  for (int c = 0; c < kRing; ++c) {
    const uint32_t goff = (uint32_t)c * (uint32_t)(kChunk * 4) + toff;
    async_copy_b128(ldsL + (uint32_t)c * kSlotBytes, rowL, goff);
    async_copy_b128(ldsN + (uint32_t)c * kSlotBytes, rowN, goff);
  }

  // Main loop: consume chunk c, then refill its slot with chunk c+kRing.
  const int mainEnd = kNChunks - kRing;  // 117
  for (int c = 0; c < mainEnd; ++c) {
    const int slot = c & (kRing - 1);
    // 2 ops/chunk, in-order completion: <=14 outstanding => chunk c landed.
    asm volatile("s_wait_asynccnt 0xe" ::: "memory");
    const float4 lz = smem.ring.L[slot][t];  // ds_load_b128, own slot only
    const float4 le = smem.ring.N[slot][t];
    // Our ds reads must land in VGPRs before the async engine may overwrite
    // the slot (async writes are not ordered against this wave's DScnt).
    asm volatile("s_wait_dscnt 0x0" ::: "memory");
    const uint32_t goff = (uint32_t)(c + kRing) * (uint32_t)(kChunk * 4) + toff;
    const uint32_t soff = (uint32_t)slot * kSlotBytes;
    async_copy_b128(ldsL + soff, rowL, goff);
    async_copy_b128(ldsN + soff, rowN, goff);

    const int base = c * kChunk + t * 4;
    PROCESS4(lz, le, base);
  }

  // Epilogue: everything outstanding has been issued long ago; drain once.
  asm volatile("s_wait_asynccnt 0x0" ::: "memory");
  for (int c = mainEnd; c < kNChunks; ++c) {
    const int slot = c & (kRing - 1);
    const float4 lz = smem.ring.L[slot][t];
    const float4 le = smem.ring.N[slot][t];
    const int base = c * kChunk + t * 4;
    PROCESS4(lz, le, base);
  }

#undef PROCESS4
#undef PROCESS_ELEM

  // Ring memory is about to be reused as reduction scratch: make sure every
  // thread is done reading its slots before anyone overwrites the union.
  __syncthreads();

  // Publish per-thread candidate sets.
  smem.red.rz [t][0] = az0; smem.red.rz [t][1] = az1; smem.red.rz [t][2] = az2;
  smem.red.rzi[t][0] = ai0; smem.red.rzi[t][1] = ai1; smem.red.rzi[t][2] = ai2;
  smem.red.rzs[t][0] = as0; smem.red.rzs[t][1] = as1; smem.red.rzs[t][2] = as2;
  smem.red.rs [t][0] = bs0; smem.red.rs [t][1] = bs1;
  smem.red.rs [t][2] = bs2; smem.red.rs [t][3] = bs3;
  smem.red.rsi[t][0] = bi0; smem.red.rsi[t][1] = bi1;
  smem.red.rsi[t][2] = bi2; smem.red.rsi[t][3] = bi3;

  // Tree reduction over the workgroup.
  for (int str = kTpb >> 1; str > 0; str >>= 1) {
    __syncthreads();
    if (t < str) {
      const int o = t + str;
      ins3(smem.red.rz[o][0], smem.red.rzi[o][0], smem.red.rzs[o][0],
           az0, ai0, as0, az1, ai1, as1, az2, ai2, as2);
      ins3(smem.red.rz[o][1], smem.red.rzi[o][1], smem.red.rzs[o][1],
           az0, ai0, as0, az1, ai1, as1, az2, ai2, as2);
      ins3(smem.red.rz[o][2], smem.red.rzi[o][2], smem.red.rzs[o][2],
           az0, ai0, as0, az1, ai1, as1, az2, ai2, as2);
      ins4(smem.red.rs[o][0], smem.red.rsi[o][0],
           bs0, bi0, bs1, bi1, bs2, bi2, bs3, bi3);
      ins4(smem.red.rs[o][1], smem.red.rsi[o][1],
           bs0, bi0, bs1, bi1, bs2, bi2, bs3, bi3);
      ins4(smem.red.rs[o][2], smem.red.rsi[o][2],
           bs0, bi0, bs1, bi1, bs2, bi2, bs3, bi3);
      ins4(smem.red.rs[o][3], smem.red.rsi[o][3],
           bs0, bi0, bs1, bi1, bs2, bi2, bs3, bi3);
      smem.red.rz [t][0] = az0; smem.red.rz [t][1] = az1; smem.red.rz [t][2] = az2;
      smem.red.rzi[t][0] = ai0; smem.red.rzi[t][1] = ai1; smem.red.rzi[t][2] = ai2;
      smem.red.rzs[t][0] = as0; smem.red.rzs[t][1] = as1; smem.red.rzs[t][2] = as2;
      smem.red.rs [t][0] = bs0; smem.red.rs [t][1] = bs1;
      smem.red.rs [t][2] = bs2; smem.red.rs [t][3] = bs3;
      smem.red.rsi[t][0] = bi0; smem.red.rsi[t][1] = bi1;
      smem.red.rsi[t][2] = bi2; smem.red.rsi[t][3] = bi3;
    }
  }

  if (t == 0) {
    int result;
    if (temp == 0.0f) {
      result = ai0;  // greedy token: argmax of original logits
    } else {
      const bool tree = (*is_tree_p) != 0;
      float best = NEG;
      int besti  = BIG;
      // Unpenalized candidates from top-4-of-s, excluding penalized positions.
      const float cs[4] = {bs0, bs1, bs2, bs3};
      const int   ci[4] = {bi0, bi1, bi2, bi3};
#pragma unroll
      for (int k = 0; k < 4; ++k) {
        const int ii = ci[k];
        if (tree && (ii == ai0 || ii == ai1 || ii == ai2)) continue;
        if (bet(cs[k], ii, best, besti)) { best = cs[k]; besti = ii; }
      }
      if (tree) {
        // penalty on logits: z' = z - phi[j]  =>  s' = s - phi[j] * invT
        const float c0 = as0 - phi[0] * invT;
        const float c1 = as1 - phi[1] * invT;
        const float c2 = as2 - phi[2] * invT;
        if (bet(c0, ai0, best, besti)) { best = c0; besti = ai0; }
        if (bet(c1, ai1, best, besti)) { best = c1; besti = ai1; }
        if (bet(c2, ai2, best, besti)) { best = c2; besti = ai2; }
      }
      result = besti;
    }
    out[row] = result;
  }
}

extern "C" void kernel_launch(void* const* d_in, const int* in_sizes, int n_in,
                              void* d_out, int out_size, void* d_ws, size_t ws_size,
                              hipStream_t stream) {
  const float* logits       = (const float*)d_in[0];
  const float* temps        = (const float*)d_in[1];
  const float* phi          = (const float*)d_in[2];
  const float* noise        = (const float*)d_in[3];
  const unsigned char* tree = (const unsigned char*)d_in[4];
  int* out = (int*)d_out;

  sampler_fused<<<dim3(kB), dim3(kTpb), 0, stream>>>(logits, temps, phi, noise, tree, out);

  (void)in_sizes; (void)n_in; (void)out_size; (void)d_ws; (void)ws_size;
}